// L1_81475529605888
// MI455X (gfx1250) — compile-verified
//
#include <hip/hip_runtime.h>
#include <math.h>

#define BATCH   8192
#define LENGTH  3000
#define K0      67      // first kept rfft bin (freq 40.2 bpm)
#define NBINS   234     // bins 67..300 inclusive
#define NPAD    256     // padded to 16 N-tiles of 16
#define NTGT    140     // interpolation targets

typedef __attribute__((ext_vector_type(2))) float v2f;
typedef __attribute__((ext_vector_type(8))) float v8f;

// ---------------------------------------------------------------------------
// Kernel 1: build cos/sin DFT tables in ONE contiguous allocation:
//   tab[n*LENGTH + k]        = cos(2*pi*((K0+n)*k mod L)/L)
//   tab[(NPAD+n)*LENGTH + k] = sin( same phase )
// Transposed [n][k] layout so WMMA B fragments are single float2 loads; sin
// rows sit at a constant +3,072,000 B offset (< 24-bit IOFFSET) from the cos
// rows so the GEMM needs only one rolling base pointer for all B fragments.
// Rows n >= NBINS are zero padding. Phase reduced exactly in integers.
// ---------------------------------------------------------------------------
__global__ void hr_gen_tables(float* __restrict__ tab) {
    const int n = blockIdx.y;
    const int k = blockIdx.x * blockDim.x + threadIdx.x;
    if (k >= LENGTH) return;
    float c = 0.0f, s = 0.0f;
    if (n < NBINS) {
        const int fi = K0 + n;
        const int ph = (fi * k) % LENGTH;   // exact integer phase reduction
        const float ang = (float)ph * (6.28318530717958647692f / (float)LENGTH);
        sincosf(ang, &s, &c);
    }
    tab[(size_t)n * LENGTH + k]          = c;
    tab[(size_t)(NPAD + n) * LENGTH + k] = s;
}

// ---------------------------------------------------------------------------
// Kernel 2: DFT-as-GEMM via V_WMMA_F32_16X16X4_F32, fused PSD->interp->Huber.
// Block = 128 threads (4 waves). Block owns 16 rows; wave w owns N-tiles
// [4w, 4w+4) -> 64 bins; block covers all 256 padded bins.
//
// Two-phase double-buffered K loop (step 8, even trip count, hand-written
// tail): phase loads fill one fragment set while the other feeds 8 WMMAs.
// No buffer rotation assignments -> no accumulator register shuffles.
//
// A fragment (16x4 f32, 2 VGPR/lane): lane m=lane&15; lanes 0-15 hold K=0,1
// lanes 16-31 hold K=2,3  -> one float2 load from X[row][kk + 2*(lane>>4)].
// B fragment (4x16 f32): lane n=lane&15, same K split -> float2 from
// tab[n][kk + 2*(lane>>4)].
// C/D: VGPR r, lane l -> element (m = r + 8*(l>>4), n = l&15).
// ---------------------------------------------------------------------------
#define TSTRIDE (16 * LENGTH)          // floats between adjacent N-tiles
#define SINOFF  (NPAD * LENGTH)        // floats from cos row to sin row

__global__ __launch_bounds__(128)
void hr_dft_psd_loss(const float* __restrict__ X,
                     const float* __restrict__ hr_gt,
                     const float* __restrict__ tab,
                     float* __restrict__ lossBuf) {
    __shared__ float psd[16][NPAD];

    const int lane  = threadIdx.x & 31;
    const int wave  = threadIdx.x >> 5;
    const int half  = lane >> 4;      // 0: K=0,1   1: K=2,3
    const int l15   = lane & 15;
    const int mBase = blockIdx.x * 16;

    const float* aPtr = X + (size_t)(mBase + l15) * LENGTH + 2 * half;
    const float* bPtr = tab + (size_t)(wave * 64 + l15) * LENGTH + 2 * half;

    v8f accRe[4] = {};
    v8f accIm[4] = {};

    v2f aF, bcF[4], bsF[4];   // fragment set F
    v2f aG, bcG[4], bsG[4];   // fragment set G

#define LOAD_SET(a_, bc_, bs_, koff)                                          \
    do {                                                                      \
        a_ = *(const v2f*)(aPtr + (koff));                                    \
        _Pragma("unroll")                                                     \
        for (int t = 0; t < 4; ++t) {                                         \
            bc_[t] = *(const v2f*)(bPtr + (koff) + t * TSTRIDE);              \
            bs_[t] = *(const v2f*)(bPtr + (koff) + t * TSTRIDE + SINOFF);     \
        }                                                                     \
    } while (0)

#define COMPUTE_SET(a_, bc_, bs_)                                             \
    do {                                                                      \
        _Pragma("unroll")                                                     \
        for (int t = 0; t < 4; ++t) {                                         \
            accRe[t] = __builtin_amdgcn_wmma_f32_16x16x4_f32(                 \
                false, a_, false, bc_[t], (short)0, accRe[t], false, false);  \
            accIm[t] = __builtin_amdgcn_wmma_f32_16x16x4_f32(                 \
                false, a_, false, bs_[t], (short)0, accIm[t], false, false);  \
        }                                                                     \
    } while (0)

    // prologue: F <- step 0
    LOAD_SET(aF, bcF, bsF, 0);

    // steady state: 374 iterations, each covering two 4-K steps
    for (int kk = 0; kk < LENGTH - 8; kk += 8) {
        LOAD_SET(aG, bcG, bsG, kk + 4);
        COMPUTE_SET(aF, bcF, bsF);
        LOAD_SET(aF, bcF, bsF, kk + 8);
        COMPUTE_SET(aG, bcG, bsG);
    }

    // tail: steps LENGTH-8 and LENGTH-4
    LOAD_SET(aG, bcG, bsG, LENGTH - 4);
    COMPUTE_SET(aF, bcF, bsF);
    COMPUTE_SET(aG, bcG, bsG);

#undef LOAD_SET
#undef COMPUTE_SET

    // PSD = Re^2 + Im^2 -> LDS (mean-centering is provably a no-op for k>=1)
#pragma unroll
    for (int t = 0; t < 4; ++t) {
        const int n = (wave * 4 + t) * 16 + l15;
#pragma unroll
        for (int r = 0; r < 8; ++r) {
            const int m = r + 8 * half;
            const float re = accRe[t][r];
            const float im = accIm[t][r];
            psd[m][n] = re * re + im * im;
        }
    }
    __syncthreads();

    // Interp to 140 targets + spectral expectation + Huber. 8 threads / row.
    const int row = threadIdx.x >> 3;   // 0..15
    const int sub = threadIdx.x & 7;    // 0..7
    float sump = 0.0f, sumwp = 0.0f;
    for (int j = sub; j < NTGT; j += 8) {
        const float tgt = 40.0f + (float)j * (140.0f / 139.0f);
        const float pos = (tgt - 40.2f) * (1.0f / 0.6f);  // uniform 0.6 bpm grid
        float p;
        if (pos <= 0.0f) {
            p = psd[row][0];                       // jnp.interp left clamp
        } else {
            int i = (int)pos;
            if (i >= NBINS - 1) {
                p = psd[row][NBINS - 1];           // right clamp (tgt==180)
            } else {
                const float fr = pos - (float)i;
                const float p0 = psd[row][i];
                p = p0 + fr * (psd[row][i + 1] - p0);
            }
        }
        sump  += p;
        sumwp += p * tgt;
    }
#pragma unroll
    for (int off = 1; off < 8; off <<= 1) {
        sump  += __shfl_xor(sump,  off, 8);
        sumwp += __shfl_xor(sumwp, off, 8);
    }
    if (sub == 0) {
        const float pred = sumwp / (sump + 1e-8f);
        const float d    = pred - hr_gt[mBase + row];
        const float ad   = fabsf(d);
        lossBuf[mBase + row] = (ad < 1.0f) ? 0.5f * d * d : ad - 0.5f;
    }
}

// ---------------------------------------------------------------------------
// Kernel 3: deterministic mean reduction of 8192 per-row losses -> d_out[0]
// ---------------------------------------------------------------------------
__global__ void hr_reduce_mean(const float* __restrict__ lossBuf,
                               float* __restrict__ out) {
    __shared__ float sm[256];
    float s = 0.0f;
    for (int i = threadIdx.x; i < BATCH; i += 256) s += lossBuf[i];
    sm[threadIdx.x] = s;
    __syncthreads();
    for (int off = 128; off > 0; off >>= 1) {
        if ((int)threadIdx.x < off) sm[threadIdx.x] += sm[threadIdx.x + off];
        __syncthreads();
    }
    if (threadIdx.x == 0) out[0] = sm[0] * (1.0f / (float)BATCH);
}

extern "C" void kernel_launch(void* const* d_in, const int* in_sizes, int n_in,
                              void* d_out, int out_size, void* d_ws, size_t ws_size,
                              hipStream_t stream) {
    const float* spr   = (const float*)d_in[0];   // [8192, 3000] f32
    const float* hr_gt = (const float*)d_in[1];   // [8192] f32

    float* tab     = (float*)d_ws;                            // 2*256*3000 f32
    float* lossBuf = tab + 2 * (size_t)NPAD * LENGTH;         // 8192 f32

    dim3 gridT((LENGTH + 255) / 256, NPAD);
    hr_gen_tables<<<gridT, 256, 0, stream>>>(tab);

    hr_dft_psd_loss<<<BATCH / 16, 128, 0, stream>>>(spr, hr_gt, tab, lossBuf);

    hr_reduce_mean<<<1, 256, 0, stream>>>(lossBuf, (float*)d_out);
}